// EPS_55379308315221
// MI455X (gfx1250) — compile-verified
//
#include <hip/hip_runtime.h>

typedef float v2f __attribute__((ext_vector_type(2)));
typedef float v4f __attribute__((ext_vector_type(4)));
typedef float v8f __attribute__((ext_vector_type(8)));

// Problem constants (from reference): K=2, C=1, S=4, O=4
// x: (1, 32, 258, 258, 4) f32 ; core: (4,4,4,4,4) f32 ; out: (32,257,257,4) f32
__global__ __launch_bounds__(256, 1)
void eps_wmma_kernel(const float* __restrict__ x,
                     const float* __restrict__ core,
                     float* __restrict__ out)
{
    constexpr int W     = 258;   // input H/W
    constexpr int WO    = 257;   // output H/W
    constexpr int TILES = 17;    // ceil(257/16) tiles of 16 pixels per row

    const int tid  = threadIdx.x;
    const int lane = tid & 31;
    const int wv   = tid >> 5;        // wave in block: 0..7
    const int n    = lane & 15;       // pixel-in-tile id / matrix column id
    const int hh   = lane >> 4;       // lane half (0/1)

    // One wave per output row (b, ho); 8 rows per block. row is wave-uniform.
    const int row = blockIdx.x * 8 + wv;        // 0 .. 32*257-1
    const int b   = row / WO;
    const int ho  = row % WO;

    const float* xrow  = x + (size_t)(b * W + ho) * W * 4;  // input row (ho)
    const float* xrow1 = xrow + (size_t)W * 4;              // input row (ho+1)
    float*       orow  = out + (size_t)row * WO * 4;

    // ---- Hoisted B operands (constant core 16x64 matrix), resident in VGPRs.
    // B layout (4x16): lane 16h+n, vgpr r holds core[(kc*4 + 2h + r)*64 + nt*16 + n]
    v2f Bm[4][4];
#pragma unroll
    for (int kc = 0; kc < 4; ++kc)
#pragma unroll
        for (int nt = 0; nt < 4; ++nt) {
            const float* bp = core + (kc * 4 + 2 * hh) * 64 + nt * 16 + n;
            Bm[kc][nt].x = bp[0];
            Bm[kc][nt].y = bp[64];
        }

    // Double-buffered T-tile spill region (per-wave private slices; no
    // inter-wave traffic, so no s_barrier is ever needed).
    __shared__ float lds[2][8][4][32][8];   // 64 KB
    const int rr = n & 7;                   // vgpr row of pixel n in D layout
    const int Lh = (n >> 3) << 4;           // lane-group base of pixel n

    // Each lane only ever uses a1[2h], a1[2h+1] -> load just that v2f (b64).
    const size_t uoff = 4 + 2 * hh;

    // ---- Prefetch tile 0 (wo = n, always valid)
    v4f a0 = *(const v4f*)(xrow  + (size_t)n * 4);
    v2f au = *(const v2f*)(xrow  + (size_t)n * 4 + uoff);
    v4f a2 = *(const v4f*)(xrow1 + (size_t)n * 4);
    v4f a3 = *(const v4f*)(xrow1 + (size_t)n * 4 + 4);

    for (int t = 0; t < TILES; ++t) {
        const int wo = t * 16 + n;

        // ---- Prefetch tile t+1 (clamped; latency hides under WMMA + epilogue,
        // since nothing below drains LOADcnt before the values are consumed).
        int wn = (t + 1) * 16 + n;
        if (wn > WO - 1) wn = WO - 1;
        v4f p0 = *(const v4f*)(xrow  + (size_t)wn * 4);
        v2f pu = *(const v2f*)(xrow  + (size_t)wn * 4 + uoff);
        v4f p2 = *(const v4f*)(xrow1 + (size_t)wn * 4);
        v4f p3 = *(const v4f*)(xrow1 + (size_t)wn * 4 + 4);

        // ---- Stage 1 A operands: u[ij] = a0[ij>>2]*a1[ij&3], 4 K-chunks.
        // A layout (16x4 f32): lane 16h+m, vgpr r holds K = 2h + r of chunk kc.
        v2f A[4];
#pragma unroll
        for (int kc = 0; kc < 4; ++kc) {
            A[kc].x = a0[kc] * au.x;
            A[kc].y = a0[kc] * au.y;
        }

        // ---- Stage 1 GEMM: T[16 pixels][64] via 16x V_WMMA_F32_16X16X4_F32
        v8f D[4];
#pragma unroll
        for (int nt = 0; nt < 4; ++nt) {
            v8f acc = {0.f, 0.f, 0.f, 0.f, 0.f, 0.f, 0.f, 0.f};
#pragma unroll
            for (int kc = 0; kc < 4; ++kc) {
                acc = __builtin_amdgcn_wmma_f32_16x16x4_f32(
                    false, A[kc], false, Bm[kc][nt], (short)0, acc, false, false);
            }
            D[nt] = acc;
        }

        // ---- Spill T tiles to LDS (D-native layout): buf[nt][lane][vgpr]
        float (*buf)[32][8] = lds[t & 1][wv];
#pragma unroll
        for (int nt = 0; nt < 4; ++nt) {
            v4f lo = { D[nt][0], D[nt][1], D[nt][2], D[nt][3] };
            v4f hi = { D[nt][4], D[nt][5], D[nt][6], D[nt][7] };
            *(v4f*)&buf[nt][lane][0] = lo;   // ds_store_b128
            *(v4f*)&buf[nt][lane][4] = hi;   // ds_store_b128
        }

        // Intra-wave cross-lane LDS RAW: LDS pipe is in-order per wave (ISA
        // §7.3); drain DScnt so the stores have landed, and pin compiler
        // ordering with a memory clobber. No s_barrier, no LOADcnt drain.
        asm volatile("s_wait_dscnt 0" ::: "memory");

        // ---- Stage 2: out[p][o] = sum_kl T[p][kl*4+o] * (a2[kl>>2]*a3[kl&3])
        // lane 16h+n handles pixel p=n, outputs o = 2h, 2h+1.
        float s0 = 0.f, s1 = 0.f;
#pragma unroll
        for (int kl = 0; kl < 16; ++kl) {
            float vv = a2[kl >> 2] * a3[kl & 3];
            int   nt = kl >> 2;
            int   c0 = (kl & 3) * 4 + 2 * hh;
            s0 += buf[nt][Lh + c0][rr] * vv;
            s1 += buf[nt][Lh + c0 + 1][rr] * vv;
        }

        if (wo < WO) {
            v2f res; res.x = s0; res.y = s1;
            *(v2f*)(orow + (size_t)wo * 4 + 2 * hh) = res;  // global_store_b64
        }

        a0 = p0; au = pu; a2 = p2; a3 = p3;   // rotate pipeline registers
    }
}

extern "C" void kernel_launch(void* const* d_in, const int* in_sizes, int n_in,
                              void* d_out, int out_size, void* d_ws, size_t ws_size,
                              hipStream_t stream) {
    const float* x    = (const float*)d_in[0];  // (1,32,258,258,4) f32
    const float* core = (const float*)d_in[1];  // (4,4,4,4,4) f32
    float* out        = (float*)d_out;          // (32,257,257,4) f32

    const int rows   = 32 * 257;   // 8224 output rows, one wave each
    const int blocks = rows / 8;   // 8 waves (rows) per 256-thread block -> 1028

    eps_wmma_kernel<<<dim3(blocks), dim3(256), 0, stream>>>(x, core, out);
}